// SegmentedPolynomial_49031346651203
// MI455X (gfx1250) — compile-verified
//
#include <hip/hip_runtime.h>

#define N_SEG   16
#define EXTENT  32
#define N_PATHS 64
#define OPSZ    512   // N_SEG * EXTENT

typedef __attribute__((ext_vector_type(2))) float v2f;
typedef __attribute__((ext_vector_type(4))) float v4f;
typedef __attribute__((ext_vector_type(8))) float v8f;

// ---------------------------------------------------------------------------
// Zero-init of out (d_out is poisoned by the harness).
// ---------------------------------------------------------------------------
__global__ void zero_f32(float* __restrict__ out, int n) {
    int i = blockIdx.x * blockDim.x + threadIdx.x;
    int i4 = i * 4;
    if (i4 + 3 < n) {
        v4f z = {0.f, 0.f, 0.f, 0.f};
        *(v4f*)(out + i4) = z;
    } else {
        for (int k = i4; k < n; ++k) out[k] = 0.f;
    }
}

// ---------------------------------------------------------------------------
// Persistent waves; one wave (32 lanes) processes one edge per iteration:
//   edge_out(16x32) = A(16x64) * T(64x32),  A[s,p] = c[p]*delta(i3[p],s),
//   T[p,k] = x1[i1[p],k] * x2[i2[p],k]
// 16 K-chunks of V_WMMA_F32_16X16X4_F32 over 2 N-tiles, then one
// global_atomic_add_f32 per output element into out[dst].
// ---------------------------------------------------------------------------
__global__ __launch_bounds__(256) void
segpoly_edge_kernel(const float* __restrict__ xn,    // (n_nodes, 512)
                    const float* __restrict__ xe,    // (n_edges, 512)
                    const float* __restrict__ coeff, // (64,)
                    const int*   __restrict__ src,   // (n_edges,)
                    const int*   __restrict__ dst,   // (n_edges,)
                    const int*   __restrict__ pidx,  // (64,3) : i1,i2,i3
                    float*       __restrict__ out,   // (n_nodes, 512)
                    int n_edges)
{
    // Per-wave LDS slices: x1 (512 f32) + x2 (512 f32), 8 waves/block.
    __shared__ float s_x[8 * 2 * OPSZ];          // 32 KB
    // Per-block B-fragment table: [hi][j][{i1(p0),i2(p0),i1(p1),i2(p1)}]
    // stored as BYTE offsets (seg*128) into the 16x32 slices.
    __shared__ int   s_off[2 * 16 * 4];          // 512 B

    const int tid  = threadIdx.x;
    const int lane = tid & 31;
    const int wib  = tid >> 5;                   // wave in block (0..7)
    const int m    = lane & 15;                  // N index within tile / M row id
    const int hi   = lane >> 4;                  // selects K rows {0,1} vs {2,3}

    // ---- build the shared index table once per block ----
    if (tid < 128) {
        int h   = tid >> 6;         // 0..1
        int r   = tid & 63;
        int j   = r >> 2;           // K-chunk 0..15
        int q   = r & 3;            // {i1(p0), i2(p0), i1(p1), i2(p1)}
        int p   = 4 * j + 2 * h + (q >> 1);
        int sel = q & 1;            // 0 -> i1, 1 -> i2
        s_off[tid] = pidx[3 * p + sel] * (EXTENT * 4);   // byte offset
    }
    __syncthreads();

    // ---- per-wave A-matrix fragments (16x4 f32 WMMA layout), built once ----
    // VGPR0: lanes 0-15 -> A[m][4j+0], lanes 16-31 -> A[m][4j+2]; VGPR1: +1.
    float a0[16], a1[16];
#pragma unroll
    for (int j = 0; j < 16; ++j) {
        int p0 = 4 * j + 2 * hi;
        a0[j] = (pidx[3 * p0 + 2] == m) ? coeff[p0]     : 0.f;
        a1[j] = (pidx[3 * p0 + 5] == m) ? coeff[p0 + 1] : 0.f;
    }

    const char* x1s = (const char*)(s_x + (wib * 2 + 0) * OPSZ);
    const char* x2s = (const char*)(s_x + (wib * 2 + 1) * OPSZ);
    v4f* v1 = (v4f*)x1s;
    v4f* v2 = (v4f*)x2s;
    const int4* offs = (const int4*)s_off;

    const int wid    = blockIdx.x * 8 + wib;     // wave-uniform
    const int nwaves = gridDim.x * 8;

#pragma unroll 1
    for (int e0 = wid; e0 < n_edges; e0 += nwaves) {
        const int e    = __builtin_amdgcn_readfirstlane(e0);
        const int srcE = __builtin_amdgcn_readfirstlane(src[e]);
        const int dstE = __builtin_amdgcn_readfirstlane(dst[e]);

        // ---- cooperative load of both rows into the wave's LDS slice ----
        {
            const v4f* pn = (const v4f*)(xn + (size_t)srcE * OPSZ);
            const v4f* pe = (const v4f*)(xe + (size_t)e    * OPSZ);
#pragma unroll
            for (int q = 0; q < 4; ++q)
                v1[lane + 32 * q] = pn[lane + 32 * q];
#pragma unroll
            for (int q = 0; q < 4; ++q)   // streamed once: keep out of L2 (NT)
                v2[lane + 32 * q] = __builtin_nontemporal_load(pe + lane + 32 * q);
        }
        // intra-wave LDS visibility: wait for this wave's DS stores, fence
        asm volatile("s_wait_dscnt 0x0" ::: "memory");
        __builtin_amdgcn_wave_barrier();

        // ---- GEMM: 16 K-chunks x 2 N-tiles of v_wmma_f32_16x16x4_f32 ----
        v8f acc0 = {0.f, 0.f, 0.f, 0.f, 0.f, 0.f, 0.f, 0.f};
        v8f acc1 = {0.f, 0.f, 0.f, 0.f, 0.f, 0.f, 0.f, 0.f};

#pragma unroll
        for (int j = 0; j < 16; ++j) {
            int4 o = offs[hi * 16 + j];  // byte offsets {i1(p0),i2(p0),i1(p1),i2(p1)}*128
            // B fragments (4x16 WMMA layout): lane supplies T[p0][col], T[p1][col]
            float b00 = *(const float*)(x1s + o.x + 4 * m) *
                        *(const float*)(x2s + o.y + 4 * m);           // tile0
            float b10 = *(const float*)(x1s + o.z + 4 * m) *
                        *(const float*)(x2s + o.w + 4 * m);
            float b01 = *(const float*)(x1s + o.x + 64 + 4 * m) *
                        *(const float*)(x2s + o.y + 64 + 4 * m);      // tile1
            float b11 = *(const float*)(x1s + o.z + 64 + 4 * m) *
                        *(const float*)(x2s + o.w + 64 + 4 * m);

            v2f A  = {a0[j], a1[j]};
            v2f B0 = {b00, b10};
            v2f B1 = {b01, b11};
            acc0 = __builtin_amdgcn_wmma_f32_16x16x4_f32(
                       false, A, false, B0, (short)0, acc0, false, false);
            acc1 = __builtin_amdgcn_wmma_f32_16x16x4_f32(
                       false, A, false, B1, (short)0, acc1, false, false);
        }

        // ---- scatter: D[M][N], M = r + 8*hi, N = m ----
        float* orow = out + (size_t)dstE * OPSZ;
#pragma unroll
        for (int r = 0; r < 8; ++r) {
            int M = r + 8 * hi;
            __hip_atomic_fetch_add(orow + M * EXTENT + m, acc0[r],
                                   __ATOMIC_RELAXED, __HIP_MEMORY_SCOPE_AGENT);
            __hip_atomic_fetch_add(orow + M * EXTENT + 16 + m, acc1[r],
                                   __ATOMIC_RELAXED, __HIP_MEMORY_SCOPE_AGENT);
        }
    }
}

// ---------------------------------------------------------------------------
extern "C" void kernel_launch(void* const* d_in, const int* in_sizes, int n_in,
                              void* d_out, int out_size, void* d_ws, size_t ws_size,
                              hipStream_t stream) {
    const float* xn    = (const float*)d_in[0];   // x_nodes
    const float* xe    = (const float*)d_in[1];   // x_edges
    const float* coeff = (const float*)d_in[2];   // path_coeffs
    const int*   src   = (const int*)d_in[3];
    const int*   dst   = (const int*)d_in[4];
    const int*   pidx  = (const int*)d_in[5];     // (64,3)
    float*       out   = (float*)d_out;

    const int n_edges = in_sizes[1] / OPSZ;

    // zero the (poisoned) output
    int nz = (out_size + 3) / 4;
    zero_f32<<<(nz + 255) / 256, 256, 0, stream>>>(out, out_size);

    // persistent waves: 2048 blocks x 8 waves = 16384 waves (~12 edges each)
    int nblocks = 2048;
    int maxblocks = (n_edges + 7) / 8;
    if (nblocks > maxblocks) nblocks = maxblocks;
    segpoly_edge_kernel<<<nblocks, 256, 0, stream>>>(
        xn, xe, coeff, src, dst, pidx, out, n_edges);
}